// SemGraphConv_29832842838122
// MI455X (gfx1250) — compile-verified
//
#include <hip/hip_runtime.h>

typedef __attribute__((ext_vector_type(2))) float v2f;
typedef __attribute__((ext_vector_type(8))) float v8f;

#define N_NODES 50000
#define N_EDGES 800000
#define F_IN    128
#define F_OUT   64

// ---------------------------------------------------------------------------
// Kernel 1: fold the two projection matrices: wsum = W[0] + W[1]
// ---------------------------------------------------------------------------
__global__ void wsum_kernel(const float* __restrict__ w, float* __restrict__ wsum) {
    int i = blockIdx.x * blockDim.x + threadIdx.x;
    if (i < F_IN * F_OUT) wsum[i] = w[i] + w[F_IN * F_OUT + i];
}

// ---------------------------------------------------------------------------
// Kernel 2: hsum = h @ wsum  via V_WMMA_F32_16X16X4_F32 (full f32 precision).
// One wave32 per 16x16 output tile; K=128 -> 32 WMMA steps.
//
// Fragment layouts (CDNA5 ISA §7.12.2, 32-bit data, wave32):
//   A (16x4):  lane l holds row M = l%16; VGPR j holds K = j + 2*(l>=16)
//   B (4x16):  lane l holds col N = l%16; VGPR j holds K = j + 2*(l>=16)
//   C/D (16x16): lane l holds col N = l%16; VGPR r holds M = r + 8*(l>=16)
// ---------------------------------------------------------------------------
__global__ void gemm_wmma_kernel(const float* __restrict__ h,
                                 const float* __restrict__ wsum,
                                 float* __restrict__ hsum) {
    const int wave = (blockIdx.x * blockDim.x + threadIdx.x) >> 5;
    const int lane = threadIdx.x & 31;
    const int ntileN = F_OUT / 16;                 // 4
    const int m0 = (wave / ntileN) * 16;
    const int n0 = (wave % ntileN) * 16;
    if (m0 >= N_NODES) return;

    const int half = lane >> 4;                    // 0: lanes 0-15, 1: lanes 16-31
    const int l16  = lane & 15;

    const float* arow = h + (m0 + l16) * F_IN + 2 * half;
    const float* bcol = wsum + n0 + l16 + 2 * half * F_OUT;

    v8f acc = {};
    #pragma unroll 4
    for (int k = 0; k < F_IN; k += 4) {
        v2f a, b;
        a.x = arow[k + 0];
        a.y = arow[k + 1];
        b.x = bcol[(k + 0) * F_OUT];
        b.y = bcol[(k + 1) * F_OUT];
        acc = __builtin_amdgcn_wmma_f32_16x16x4_f32(
            /*neg_a=*/false, a, /*neg_b=*/false, b,
            /*c_mod=*/(short)0, acc, /*reuse_a=*/false, /*reuse_b=*/false);
    }

    float* drow = hsum + m0 * F_OUT + n0 + l16;
    #pragma unroll
    for (int r = 0; r < 8; ++r)
        drow[(r + 8 * half) * F_OUT] = acc[r];
}

// ---------------------------------------------------------------------------
// Kernel 3: out[n, c] = bias[c]  (bias applied once per node; edges accumulate)
// ---------------------------------------------------------------------------
__global__ void init_out_kernel(const float* __restrict__ bias, float* __restrict__ out) {
    int i = blockIdx.x * blockDim.x + threadIdx.x;
    if (i < N_NODES * F_OUT) out[i] = bias[i & (F_OUT - 1)];
}

// ---------------------------------------------------------------------------
// Kernel 4: per-edge softmax + gather + scatter-add.
// One wave32 per edge; each lane owns columns {lane, lane+32}.
// edge_feat streamed once (204.8 MB); hsum gathers hit L2 (12.8 MB table);
// atomics lower to global_atomic_add_f32 (no-return).
// ---------------------------------------------------------------------------
__global__ void edge_kernel(const float* __restrict__ edge_feat,
                            const int*   __restrict__ src,
                            const int*   __restrict__ dst,
                            const float* __restrict__ hsum,
                            float*       __restrict__ out) {
    const int wave = (blockIdx.x * (blockDim.x >> 5)) + (threadIdx.x >> 5);
    const int lane = threadIdx.x & 31;
    if (wave >= N_EDGES) return;

    const float f0 = edge_feat[wave * F_OUT + lane];
    const float f1 = edge_feat[wave * F_OUT + lane + 32];

    // softmax over the 64-wide feature axis (wave32 tree reduction)
    float mx = fmaxf(f0, f1);
    #pragma unroll
    for (int off = 16; off > 0; off >>= 1)
        mx = fmaxf(mx, __shfl_xor(mx, off, 32));

    const float e0 = __expf(f0 - mx);
    const float e1 = __expf(f1 - mx);
    float s = e0 + e1;
    #pragma unroll
    for (int off = 16; off > 0; off >>= 1)
        s += __shfl_xor(s, off, 32);
    const float inv = 1.0f / s;

    const int sn = src[wave];
    const int dn = dst[wave];
    const float g0 = hsum[sn * F_OUT + lane];
    const float g1 = hsum[sn * F_OUT + lane + 32];

    atomicAdd(&out[dn * F_OUT + lane],      g0 * e0 * inv);
    atomicAdd(&out[dn * F_OUT + lane + 32], g1 * e1 * inv);
}

// ---------------------------------------------------------------------------
extern "C" void kernel_launch(void* const* d_in, const int* in_sizes, int n_in,
                              void* d_out, int out_size, void* d_ws, size_t ws_size,
                              hipStream_t stream) {
    const float* h         = (const float*)d_in[0];
    const float* edge_feat = (const float*)d_in[1];
    const float* weight    = (const float*)d_in[2];
    const float* bias      = (const float*)d_in[3];
    const int*   src       = (const int*)d_in[4];
    const int*   dst       = (const int*)d_in[5];
    float* out = (float*)d_out;

    // workspace layout: [ wsum: 128*64 f32 | hsum: 50000*64 f32 ]
    float* wsum = (float*)d_ws;
    float* hsum = (float*)((char*)d_ws + ((F_IN * F_OUT * sizeof(float) + 255u) & ~255u));

    // 1) fold weights
    wsum_kernel<<<(F_IN * F_OUT + 255) / 256, 256, 0, stream>>>(weight, wsum);

    // 2) WMMA GEMM: 3125 * 4 = 12500 tiles, 8 waves (256 threads) per block
    const int ntiles = (N_NODES / 16) * (F_OUT / 16);
    gemm_wmma_kernel<<<(ntiles + 7) / 8, 256, 0, stream>>>(h, wsum, hsum);

    // 3) seed output with bias
    init_out_kernel<<<(N_NODES * F_OUT + 255) / 256, 256, 0, stream>>>(bias, out);

    // 4) edge softmax + gather + atomic scatter-add (1 wave per edge)
    edge_kernel<<<(N_EDGES + 7) / 8, 256, 0, stream>>>(edge_feat, src, dst, hsum, out);
}